// HMPNNLayer_19327352832462
// MI455X (gfx1250) — compile-verified
//
#include <hip/hip_runtime.h>

#define F 128
#define BN_EPS 1e-5f

typedef __attribute__((ext_vector_type(2))) float v2f;
typedef __attribute__((ext_vector_type(8))) float v8f;

__device__ __forceinline__ float fast_sigmoid(float x) {
    return 1.0f / (1.0f + __expf(-x));
}

// Guaranteed hardware fp32 atomic add (no-return form, GV addressing:
// 64-bit VGPR-pair address, SADDR=off). RMW executes at the owning cache
// level (L2 here, since the agg buffers are L2-resident) and is tracked
// with STOREcnt; the implicit wait at s_endpgm covers completion.
__device__ __forceinline__ void global_atomic_fadd(float* p, float v) {
    asm volatile("global_atomic_add_f32 %0, %1, off"
                 :: "v"((unsigned long long)p), "v"(v)
                 : "memory");
}

// ---------------------------------------------------------------------------
// Zero a float buffer (float4 stores). n must be a multiple of 4.
// ---------------------------------------------------------------------------
__global__ void zero_f32x4(float4* __restrict__ p, long long n4) {
    long long i = (long long)blockIdx.x * blockDim.x + threadIdx.x;
    if (i < n4) p[i] = make_float4(0.f, 0.f, 0.f, 0.f);
}

// ---------------------------------------------------------------------------
// GEMM1: out[m,n] = sigmoid( sum_k A[m,k] * W[n,k] + bias[n] )
//   A:[M,128], W:[128,128] row-major (out_feat, in_feat), N=K=128.
// One wave per 16-row tile; wave iterates all 8 N-tiles (A frag reused 8x).
// f32 WMMA 16x16x4: A frag = float2 {A[laneM, k+2*hi], A[laneM, k+2*hi+1]}
//                   B frag = float2 {W[n,   k+2*hi], W[n,   k+2*hi+1]}
// C/D layout: vgpr v, lane half h -> row = v + 8h, col = lane&15.
// ---------------------------------------------------------------------------
__global__ void __launch_bounds__(256) gemm_sig_nn(
    const float* __restrict__ A, const float* __restrict__ W,
    const float* __restrict__ bias, float* __restrict__ out, int M)
{
    const int wave = threadIdx.x >> 5;
    const int lane = threadIdx.x & 31;
    const int hi   = lane >> 4;        // 0 or 1
    const int lm   = lane & 15;

    const int mTile = blockIdx.x * 8 + wave;
    if (mTile * 16 >= M) return;       // wave-uniform: EXEC stays all-ones
    const int row0 = mTile * 16;

    v8f acc[8] = {};
    const float* aRow = A + (size_t)(row0 + lm) * F;

    for (int k = 0; k < F; k += 4) {
        const int ka = k + 2 * hi;
        const v2f a = *(const v2f*)(aRow + ka);
        #pragma unroll
        for (int nt = 0; nt < 8; ++nt) {
            const v2f b = *(const v2f*)(W + (size_t)(nt * 16 + lm) * F + ka);
            acc[nt] = __builtin_amdgcn_wmma_f32_16x16x4_f32(
                false, a, false, b, (short)0, acc[nt], false, false);
        }
    }

    #pragma unroll
    for (int nt = 0; nt < 8; ++nt) {
        const int c  = nt * 16 + lm;
        const float bs = bias[c];
        #pragma unroll
        for (int v = 0; v < 8; ++v) {
            const int r = row0 + v + 8 * hi;
            out[(size_t)r * F + c] = fast_sigmoid(acc[nt][v] + bs);
        }
    }
}

// ---------------------------------------------------------------------------
// GEMM2: out[m,n] = sigmoid( sum_{k<256} cat(A0,A1)[m,k] * W[n,k] + bias[n] )
//   A0,A1:[M,128] (virtual concat along k), W:[128,256], N=128, K=256.
// ---------------------------------------------------------------------------
__global__ void __launch_bounds__(256) gemm_sig_cat(
    const float* __restrict__ A0, const float* __restrict__ A1,
    const float* __restrict__ W, const float* __restrict__ bias,
    float* __restrict__ out, int M)
{
    const int wave = threadIdx.x >> 5;
    const int lane = threadIdx.x & 31;
    const int hi   = lane >> 4;
    const int lm   = lane & 15;

    const int mTile = blockIdx.x * 8 + wave;
    if (mTile * 16 >= M) return;
    const int row0 = mTile * 16;

    v8f acc[8] = {};
    const float* a0Row = A0 + (size_t)(row0 + lm) * F;
    const float* a1Row = A1 + (size_t)(row0 + lm) * F;

    for (int k = 0; k < 2 * F; k += 4) {
        const int ka = k + 2 * hi;
        // k-step of 4 with hi*2 offset never crosses the 128 boundary
        const v2f a = (k < F) ? *(const v2f*)(a0Row + ka)
                              : *(const v2f*)(a1Row + (ka - F));
        #pragma unroll
        for (int nt = 0; nt < 8; ++nt) {
            const v2f b = *(const v2f*)(W + (size_t)(nt * 16 + lm) * (2 * F) + ka);
            acc[nt] = __builtin_amdgcn_wmma_f32_16x16x4_f32(
                false, a, false, b, (short)0, acc[nt], false, false);
        }
    }

    #pragma unroll
    for (int nt = 0; nt < 8; ++nt) {
        const int c  = nt * 16 + lm;
        const float bs = bias[c];
        #pragma unroll
        for (int v = 0; v < 8; ++v) {
            const int r = row0 + v + 8 * hi;
            out[(size_t)r * F + c] = fast_sigmoid(acc[nt][v] + bs);
        }
    }
}

// ---------------------------------------------------------------------------
// Scatter-sum: for each incidence pair p: dst[sidx[p], :] += src[gidx[p], :]
// One 32-lane wave per pair; each lane handles 4 floats (float4 gather,
// 4 global_atomic_add_f32 -> L2-resident RMW, no CAS loop).
// ---------------------------------------------------------------------------
__global__ void __launch_bounds__(256) scatter_add_rows(
    const float* __restrict__ src, const int* __restrict__ gidx,
    const int* __restrict__ sidx, float* __restrict__ dst, int nnz)
{
    const long long t = (long long)blockIdx.x * blockDim.x + threadIdx.x;
    const long long pair = t >> 5;
    const int lane = (int)(t & 31);
    if (pair >= nnz) return;

    const int g = gidx[pair];
    const int s = sidx[pair];
    const float4 v = *(const float4*)(src + (size_t)g * F + lane * 4);
    float* d = dst + (size_t)s * F + lane * 4;
    global_atomic_fadd(d + 0, v.x);
    global_atomic_fadd(d + 1, v.y);
    global_atomic_fadd(d + 2, v.z);
    global_atomic_fadd(d + 3, v.w);
}

// ---------------------------------------------------------------------------
// Finalize: out = sigmoid( (x - mean) * (gamma * rsqrt(var+eps)) + beta + agg )
// ---------------------------------------------------------------------------
__global__ void __launch_bounds__(256) bn_add_sig(
    const float* __restrict__ x, const float* __restrict__ agg,
    const float* __restrict__ gamma, const float* __restrict__ beta,
    const float* __restrict__ mean, const float* __restrict__ var,
    float* __restrict__ out, long long n)
{
    const long long i = (long long)blockIdx.x * blockDim.x + threadIdx.x;
    if (i >= n) return;
    const int c = (int)(i & (F - 1));
    const float scale = gamma[c] * rsqrtf(var[c] + BN_EPS);
    const float v = (x[i] - mean[c]) * scale + beta[c] + agg[i];
    out[i] = fast_sigmoid(v);
}

// ---------------------------------------------------------------------------
extern "C" void kernel_launch(void* const* d_in, const int* in_sizes, int n_in,
                              void* d_out, int out_size, void* d_ws, size_t ws_size,
                              hipStream_t stream)
{
    const float* x0       = (const float*)d_in[0];
    const float* x1       = (const float*)d_in[1];
    const int*   node_idx = (const int*)d_in[2];
    const int*   edge_idx = (const int*)d_in[3];
    const float* W_nh     = (const float*)d_in[4];
    const float* b_nh     = (const float*)d_in[5];
    const float* W_en     = (const float*)d_in[6];
    const float* b_en     = (const float*)d_in[7];
    const float* bn0_g = (const float*)d_in[8];
    const float* bn0_b = (const float*)d_in[9];
    const float* bn0_m = (const float*)d_in[10];
    const float* bn0_v = (const float*)d_in[11];
    const float* bn1_g = (const float*)d_in[12];
    const float* bn1_b = (const float*)d_in[13];
    const float* bn1_m = (const float*)d_in[14];
    const float* bn1_v = (const float*)d_in[15];

    const int n_nodes = in_sizes[0] / F;
    const int n_edges = in_sizes[1] / F;
    const int nnz     = in_sizes[2];

    // Workspace layout (he_msg_agg and node_msg_agg adjacent so they can be
    // zeroed in one pass):
    float* node_messages = (float*)d_ws;                              // [n_nodes,F]
    float* he_msg_agg    = node_messages + (size_t)n_nodes * F;       // [n_nodes,F]
    float* node_msg_agg  = he_msg_agg    + (size_t)n_nodes * F;       // [n_edges,F]
    float* he_messages   = node_msg_agg  + (size_t)n_edges * F;       // [n_edges,F]

    float* out_x0 = (float*)d_out;                    // [n_nodes,F]
    float* out_x1 = out_x0 + (size_t)n_nodes * F;     // [n_edges,F]

    // 1) zero both accumulator buffers
    {
        const long long n4 = ((long long)(n_nodes + n_edges) * F) / 4;
        const int blocks = (int)((n4 + 255) / 256);
        zero_f32x4<<<blocks, 256, 0, stream>>>((float4*)he_msg_agg, n4);
    }

    // 2) node messenger GEMM: node_messages = sigmoid(x0 @ W_nh^T + b_nh)
    {
        const int mTiles = (n_nodes + 15) / 16;
        const int blocks = (mTiles + 7) / 8;
        gemm_sig_nn<<<blocks, 256, 0, stream>>>(x0, W_nh, b_nh, node_messages, n_nodes);
    }

    // 3) scatter node messages onto hyperedges
    {
        const long long threads = (long long)nnz * 32;
        const int blocks = (int)((threads + 255) / 256);
        scatter_add_rows<<<blocks, 256, 0, stream>>>(
            node_messages, node_idx, edge_idx, node_msg_agg, nnz);
    }

    // 4) hyperedge messenger GEMM: he_messages = sigmoid(cat(x1,agg) @ W_en^T + b_en)
    {
        const int mTiles = (n_edges + 15) / 16;
        const int blocks = (mTiles + 7) / 8;
        gemm_sig_cat<<<blocks, 256, 0, stream>>>(
            x1, node_msg_agg, W_en, b_en, he_messages, n_edges);
    }

    // 5) scatter hyperedge messages onto nodes
    {
        const long long threads = (long long)nnz * 32;
        const int blocks = (int)((threads + 255) / 256);
        scatter_add_rows<<<blocks, 256, 0, stream>>>(
            he_messages, edge_idx, node_idx, he_msg_agg, nnz);
    }

    // 6) finalize
    {
        const long long n0 = (long long)n_nodes * F;
        bn_add_sig<<<(int)((n0 + 255) / 256), 256, 0, stream>>>(
            x0, he_msg_agg, bn0_g, bn0_b, bn0_m, bn0_v, out_x0, n0);
        const long long n1 = (long long)n_edges * F;
        bn_add_sig<<<(int)((n1 + 255) / 256), 256, 0, stream>>>(
            x1, node_msg_agg, bn1_g, bn1_b, bn1_m, bn1_v, out_x1, n1);
    }
}